// SNN_49340584296534
// MI455X (gfx1250) — compile-verified
//
#include <hip/hip_runtime.h>
#include <math.h>

typedef float v2f __attribute__((ext_vector_type(2)));
typedef float v8f __attribute__((ext_vector_type(8)));

#define N_NODES 100000
#define D_STALK 4
#define H_CH    64
#define IN_CH   256
#define OUT_CH  64
#define L_LAYERS 4
#define NNZ_CNT 8000000
#define ND      (N_NODES * D_STALK)

__device__ __forceinline__ float elu1(float v) {
    return v > 0.0f ? v : (__expf(v) - 1.0f);
}

__device__ __forceinline__ void atomic_add_f32(float* p, float v) {
    __hip_atomic_fetch_add(p, v, __ATOMIC_RELAXED, __HIP_MEMORY_SCOPE_AGENT);
}

// ---------------------------------------------------------------------------
// lin1: h = elu(x @ W1^T + b1), x:[N,256], W1:[256,256] (torch [out,in]).
// Output buffer [N,256] == reshaped xc [N*D, 64] (identical row-major layout).
// Block: 512 threads = 16 waves; each wave owns one 16-wide column tile.
// Dual accumulators break the serial WMMA dependency chain.
// ---------------------------------------------------------------------------
__global__ __launch_bounds__(512) void k_lin1(const float* __restrict__ x,
                                              const float* __restrict__ W1,
                                              const float* __restrict__ b1,
                                              float* __restrict__ out) {
    __shared__ float xs[16][258];   // 258 stride: avoid 64-bank conflicts
    const int tid = threadIdx.x;
    const int rowBase = blockIdx.x * 16;

    // stage 16x256 rows of x into LDS (float4, coalesced)
    for (int idx = tid; idx < 16 * 64; idx += 512) {
        const int r  = idx >> 6;
        const int c4 = (idx & 63) << 2;
        const float4 v = *(const float4*)(x + (size_t)(rowBase + r) * IN_CH + c4);
        xs[r][c4 + 0] = v.x; xs[r][c4 + 1] = v.y;
        xs[r][c4 + 2] = v.z; xs[r][c4 + 3] = v.w;
    }
    __syncthreads();

    const int wave = tid >> 5;
    const int lane = tid & 31;
    const int half = lane >> 4;        // 0: lanes 0-15, 1: lanes 16-31
    const int l16  = lane & 15;
    const int koff = half * 2;
    const int colBase = wave * 16;

    v8f acc0 = {}, acc1 = {};
    for (int k = 0; k < IN_CH; k += 8) {
        v2f a0, b0, a1, b1v;
        a0.x = xs[l16][k + koff];
        a0.y = xs[l16][k + koff + 1];
        a1.x = xs[l16][k + 4 + koff];
        a1.y = xs[l16][k + 4 + koff + 1];
        const float* wrow = W1 + (size_t)(colBase + l16) * IN_CH + k + koff;
        b0.x = wrow[0];
        b0.y = wrow[1];
        b1v.x = wrow[4];
        b1v.y = wrow[5];
        acc0 = __builtin_amdgcn_wmma_f32_16x16x4_f32(false, a0, false, b0,
                                                     (short)0, acc0, false, false);
        acc1 = __builtin_amdgcn_wmma_f32_16x16x4_f32(false, a1, false, b1v,
                                                     (short)0, acc1, false, false);
    }
    const v8f acc = acc0 + acc1;

    const float bias = b1[colBase + l16];
#pragma unroll
    for (int i = 0; i < 8; ++i) {
        const int row = rowBase + i + half * 8;
        out[(size_t)row * IN_CH + colBase + l16] = elu1(acc[i] + bias);
    }
}

// ---------------------------------------------------------------------------
// Per layer: y = (blockdiag(Wl) @ xc) @ Wr^T  over 32-row (8-node) blocks.
// Block: 256 threads = 8 waves; waves tile the 32x64 output (2 M x 4 N tiles).
// ---------------------------------------------------------------------------
__global__ __launch_bounds__(256) void k_leftright(const float* __restrict__ xc,
                                                   const float* __restrict__ Wl,
                                                   const float* __restrict__ Wr,
                                                   float* __restrict__ y) {
    __shared__ float xs[32][66];
    __shared__ float ts[32][66];
    const int tid = threadIdx.x;
    const int rowBase = blockIdx.x * 32;

    // stage 32x64 activation block
    for (int idx = tid; idx < 32 * 16; idx += 256) {
        const int r  = idx >> 4;
        const int c4 = (idx & 15) << 2;
        const float4 v = *(const float4*)(xc + (size_t)(rowBase + r) * H_CH + c4);
        xs[r][c4 + 0] = v.x; xs[r][c4 + 1] = v.y;
        xs[r][c4 + 2] = v.z; xs[r][c4 + 3] = v.w;
    }
    float wl[16];
#pragma unroll
    for (int i = 0; i < 16; ++i) wl[i] = Wl[i];   // uniform -> scalar loads
    __syncthreads();

    // left multiply: t[r][c] = sum_d' Wl[r%4][d'] * xs[(r&~3)+d'][c]
    for (int idx = tid; idx < 32 * 64; idx += 256) {
        const int r = idx >> 6, c = idx & 63;
        const int nb = r & ~3, d = (r & 3) * 4;
        ts[r][c] = wl[d + 0] * xs[nb + 0][c] + wl[d + 1] * xs[nb + 1][c]
                 + wl[d + 2] * xs[nb + 2][c] + wl[d + 3] * xs[nb + 3][c];
    }
    __syncthreads();

    const int wave = tid >> 5;
    const int lane = tid & 31;
    const int half = lane >> 4;
    const int l16  = lane & 15;
    const int koff = half * 2;
    const int mBase   = (wave >> 2) * 16;
    const int colBase = (wave & 3) * 16;

    v8f acc0 = {}, acc1 = {};
    for (int k = 0; k < H_CH; k += 8) {
        v2f a0, b0, a1, b1v;
        a0.x = ts[mBase + l16][k + koff];
        a0.y = ts[mBase + l16][k + koff + 1];
        a1.x = ts[mBase + l16][k + 4 + koff];
        a1.y = ts[mBase + l16][k + 4 + koff + 1];
        const float* wr = Wr + (size_t)(colBase + l16) * H_CH + k + koff;
        b0.x = wr[0];
        b0.y = wr[1];
        b1v.x = wr[4];
        b1v.y = wr[5];
        acc0 = __builtin_amdgcn_wmma_f32_16x16x4_f32(false, a0, false, b0,
                                                     (short)0, acc0, false, false);
        acc1 = __builtin_amdgcn_wmma_f32_16x16x4_f32(false, a1, false, b1v,
                                                     (short)0, acc1, false, false);
    }
    const v8f acc = acc0 + acc1;
#pragma unroll
    for (int i = 0; i < 8; ++i) {
        const int row = rowBase + mBase + i + half * 8;
        y[(size_t)row * H_CH + colBase + l16] = acc[i];
    }
}

// ---------------------------------------------------------------------------
// SpMM scatter: out[rows[i], :] += vals[i] * y[cols[i], :]
// One wave per 32 nnz: coalesced metadata loads (one lane each), then
// per-nnz broadcast via v_readlane. Gather rows prefetched up front
// (global_prefetch_b8, per-lane addresses) so the serial gather loop hits
// cache. Each lane covers 2 of the 64 channels; y/out (~102 MB) are
// L2-resident on the 192 MB L2; scatter uses agent-scope f32 atomics.
// ---------------------------------------------------------------------------
__global__ __launch_bounds__(256) void k_spmm(const float* __restrict__ y,
                                              const float* __restrict__ vals,
                                              const int* __restrict__ rows,
                                              const int* __restrict__ cols,
                                              float* __restrict__ out) {
    const int lane = threadIdx.x & 31;
    const int base = (blockIdx.x * 8 + (threadIdx.x >> 5)) * 32;

    const int   r = rows[base + lane];   // coalesced
    const int   c = cols[base + lane];   // coalesced
    const float v = vals[base + lane];   // coalesced

    // prefetch this wave's 32 gather rows (256 B each -> two cachelines)
    const float* yrow = y + (size_t)c * H_CH;
    __builtin_prefetch(yrow, 0, 1);
    __builtin_prefetch(yrow + 32, 0, 1);

#pragma unroll
    for (int j = 0; j < 32; ++j) {
        const int   rj = __shfl(r, j);
        const int   cj = __shfl(c, j);
        const float vj = __shfl(v, j);
        const float2 s = *(const float2*)(y + (size_t)cj * H_CH + lane * 2);
        float* dst = out + (size_t)rj * H_CH + lane * 2;
        atomic_add_f32(dst + 0, vj * s.x);
        atomic_add_f32(dst + 1, vj * s.y);
    }
}

// ---------------------------------------------------------------------------
// Pointwise: x0 = (1 + tanh(eps[l][d])) * x0 - elu(spmm); also re-zero spmm
// accumulator for the next layer. float4 over ND*H elements.
// ---------------------------------------------------------------------------
__global__ __launch_bounds__(256) void k_update(float* __restrict__ x0,
                                                float* __restrict__ spmm,
                                                const float* __restrict__ eps,
                                                int layer) {
    const size_t idx4 = (size_t)blockIdx.x * 256 + threadIdx.x;
    const int row = (int)((idx4 * 4) >> 6);     // ND-row index
    const int d = row & 3;
    const float coeff = 1.0f + tanhf(eps[layer * 4 + d]);

    float4 v = ((float4*)spmm)[idx4];
    const float4 z = {0.0f, 0.0f, 0.0f, 0.0f};
    ((float4*)spmm)[idx4] = z;

    float4 a = ((float4*)x0)[idx4];
    a.x = coeff * a.x - elu1(v.x);
    a.y = coeff * a.y - elu1(v.y);
    a.z = coeff * a.z - elu1(v.z);
    a.w = coeff * a.w - elu1(v.w);
    ((float4*)x0)[idx4] = a;
}

__global__ __launch_bounds__(256) void k_zero(float4* __restrict__ p) {
    const float4 z = {0.0f, 0.0f, 0.0f, 0.0f};
    p[(size_t)blockIdx.x * 256 + threadIdx.x] = z;
}

// ---------------------------------------------------------------------------
// lin2: out = xc.reshape(N,256) @ W2^T + b2 ; block: 128 thr = 4 waves.
// ---------------------------------------------------------------------------
__global__ __launch_bounds__(128) void k_lin2(const float* __restrict__ xc,
                                              const float* __restrict__ W2,
                                              const float* __restrict__ b2,
                                              float* __restrict__ out) {
    __shared__ float xs[16][258];
    const int tid = threadIdx.x;
    const int rowBase = blockIdx.x * 16;

    for (int idx = tid; idx < 16 * 64; idx += 128) {
        const int r  = idx >> 6;
        const int c4 = (idx & 63) << 2;
        const float4 v = *(const float4*)(xc + (size_t)(rowBase + r) * IN_CH + c4);
        xs[r][c4 + 0] = v.x; xs[r][c4 + 1] = v.y;
        xs[r][c4 + 2] = v.z; xs[r][c4 + 3] = v.w;
    }
    __syncthreads();

    const int wave = tid >> 5;
    const int lane = tid & 31;
    const int half = lane >> 4;
    const int l16  = lane & 15;
    const int koff = half * 2;
    const int colBase = wave * 16;

    v8f acc0 = {}, acc1 = {};
    for (int k = 0; k < IN_CH; k += 8) {
        v2f a0, b0, a1, b1v;
        a0.x = xs[l16][k + koff];
        a0.y = xs[l16][k + koff + 1];
        a1.x = xs[l16][k + 4 + koff];
        a1.y = xs[l16][k + 4 + koff + 1];
        const float* w = W2 + (size_t)(colBase + l16) * IN_CH + k + koff;
        b0.x = w[0];
        b0.y = w[1];
        b1v.x = w[4];
        b1v.y = w[5];
        acc0 = __builtin_amdgcn_wmma_f32_16x16x4_f32(false, a0, false, b0,
                                                     (short)0, acc0, false, false);
        acc1 = __builtin_amdgcn_wmma_f32_16x16x4_f32(false, a1, false, b1v,
                                                     (short)0, acc1, false, false);
    }
    const v8f acc = acc0 + acc1;
    const float bias = b2[colBase + l16];
#pragma unroll
    for (int i = 0; i < 8; ++i) {
        const int row = rowBase + i + half * 8;
        out[(size_t)row * OUT_CH + colBase + l16] = acc[i] + bias;
    }
}

extern "C" void kernel_launch(void* const* d_in, const int* in_sizes, int n_in,
                              void* d_out, int out_size, void* d_ws, size_t ws_size,
                              hipStream_t stream) {
    const float* x    = (const float*)d_in[0];
    const float* vals = (const float*)d_in[1];
    const float* W1   = (const float*)d_in[2];
    const float* b1   = (const float*)d_in[3];
    const float* Wl   = (const float*)d_in[4];   // [L,4,4]
    const float* Wr   = (const float*)d_in[5];   // [L,64,64]
    const float* eps  = (const float*)d_in[6];   // [L,4,1]
    const float* W2   = (const float*)d_in[7];
    const float* b2   = (const float*)d_in[8];
    const int* rows   = (const int*)d_in[9];
    const int* cols   = (const int*)d_in[10];
    float* out = (float*)d_out;

    const size_t bufElems = (size_t)ND * H_CH;   // 25.6M floats
    float* bufB = (float*)d_ws;                  // x0 == xc (102.4 MB)
    float* bufC = bufB + bufElems;               // left/right GEMM output
    float* bufD = bufC + bufElems;               // SpMM accumulator

    const int vec4Blocks = (int)(bufElems / 4 / 256);   // 25000

    k_zero<<<vec4Blocks, 256, 0, stream>>>((float4*)bufD);
    k_lin1<<<N_NODES / 16, 512, 0, stream>>>(x, W1, b1, bufB);

    for (int l = 0; l < L_LAYERS; ++l) {
        k_leftright<<<ND / 32, 256, 0, stream>>>(bufB, Wl + l * 16,
                                                 Wr + (size_t)l * 64 * 64, bufC);
        k_spmm<<<NNZ_CNT / 256, 256, 0, stream>>>(bufC, vals, rows, cols, bufD);
        k_update<<<vec4Blocks, 256, 0, stream>>>(bufB, bufD, eps, l);
    }

    k_lin2<<<N_NODES / 16, 128, 0, stream>>>(bufB, W2, b2, out);
}